// SpectralConv3d_9337258902439
// MI455X (gfx1250) — compile-verified
//
#include <hip/hip_runtime.h>

// SpectralConv3d on MI455X (gfx1250, wave32).
// Bandwidth floor: 512MB in/out @ 23.3 TB/s ~ 25us. Compute ~7 GFLOP via
// pruned DFTs; the two large GEMM-shaped stages (forward/inverse Z transform,
// ~2.1 GFLOP each) run on V_WMMA_F32_16X16X4_F32 to keep fp32 accuracy.
// Bulk global->LDS traffic goes through the gfx1250 async-to-LDS engine
// (ASYNCcnt) when the builtins are available.

#define PI_F 3.14159265358979323846f

typedef float v2f __attribute__((ext_vector_type(2)));
typedef float v8f __attribute__((ext_vector_type(8)));
typedef int v4i __attribute__((vector_size(16)));  // async-LDS builtin operand

#if defined(__has_builtin)
#if __has_builtin(__builtin_amdgcn_global_load_async_to_lds_b128) && \
    __has_builtin(__builtin_amdgcn_s_wait_asynccnt)
#define HAVE_ASYNC_LDS 1
#endif
#endif
#ifndef HAVE_ASYNC_LDS
#define HAVE_ASYNC_LDS 0
#endif

__device__ inline v8f wmma_f32_k4(v2f a, v2f b, v8f c) {
  // D = A(16x4 f32) * B(4x16 f32) + C(16x16 f32)
  return __builtin_amdgcn_wmma_f32_16x16x4_f32(
      /*neg_a=*/false, a, /*neg_b=*/false, b,
      /*c_mod=*/(short)0, c, /*reuse_a=*/false, /*reuse_b=*/false);
}

// ---------------------------------------------------------------------------
// K1: fused forward Z-DFT (WMMA) + forward Y-DFT (VALU) for one (b,c,d) plane.
// Input plane x[b][c][d][y][z] (64x64), output F12[b][c][d][kh16][kz8][2].
// Forward ortho factor 1/512 folded into the Z DFT matrix.
// ---------------------------------------------------------------------------
__global__ __launch_bounds__(256) void k1_fwd_zy(const float* __restrict__ x,
                                                 float* __restrict__ F12) {
  // row stride 68 floats: 16B-aligned rows for async B128 fills AND
  // conflict-free A-fragment reads (word addr m*68+kk -> 4m+kk mod 64 distinct)
  __shared__ __align__(16) float plane[64 * 68];
  __shared__ float dftz[64 * 16];    // [z][2kz+p], p=0:cos, p=1:-sin (scaled 1/512)
  __shared__ float twy[16 * 64 * 2]; // [kh_i][y] {cos,sin} of 2*pi*kh*y/64
  __shared__ float zf[64 * 17];      // Z-DFT result [y][2kz+p], padded

  const int t = threadIdx.x;
  const int blk = blockIdx.x;  // (b*32+c)*64 + d
  const float* src = x + (size_t)blk * 4096;

  // phase 0: plane fill via async global->LDS engine (16B granules; rows are
  // 64 floats so 16B chunks never cross the 68-float padded row boundary).
#if HAVE_ASYNC_LDS
  {
    v4i* g = (v4i*)(void*)const_cast<float*>(src);
#pragma unroll
    for (int i = 0; i < 4; ++i) {
      int c4 = t + i * 256;                  // 16-byte chunk id, 0..1023
      int row = c4 >> 4, col = (c4 & 15) * 4;
      __builtin_amdgcn_global_load_async_to_lds_b128(
          g + c4, (v4i*)(void*)&plane[row * 68 + col], 0, 0);
    }
    __builtin_amdgcn_s_wait_asynccnt(0);
  }
#else
#pragma unroll
  for (int i = 0; i < 16; ++i) {
    int e = t + i * 256;  // 0..4095
    plane[(e >> 6) * 68 + (e & 63)] = src[e];
  }
#endif
#pragma unroll
  for (int i = 0; i < 4; ++i) {
    int e = t * 4 + i;  // z*16 + comp
    int z = e >> 4, comp = e & 15, kz = comp >> 1;
    float th = (2.0f * PI_F / 64.0f) * (float)(kz * z);
    float sv, cv; __sincosf(th, &sv, &cv);
    dftz[e] = ((comp & 1) ? -sv : cv) * (1.0f / 512.0f);
  }
#pragma unroll
  for (int i = 0; i < 4; ++i) {
    int e = t * 4 + i;  // kh_i*64 + y
    int kh_i = e >> 6, y = e & 63;
    int kh = (kh_i < 8) ? kh_i : kh_i + 48;
    float th = (2.0f * PI_F / 64.0f) * (float)(kh * y);
    float sv, cv; __sincosf(th, &sv, &cv);
    twy[e * 2] = cv; twy[e * 2 + 1] = sv;
  }
  __syncthreads();

  // phase 1: zf[64y][16] = plane[64y][64z] * dftz[64z][16] via WMMA f32 16x16x4
  const int wave = t >> 5, lane = t & 31;
  const int ln15 = lane & 15;
  const int khalf = (lane >> 4) * 2;  // K offset for this half-wave
  if (wave < 4) {                     // wave-uniform branch, EXEC all ones inside
    const int yt = wave;              // y-tile 0..3
    v8f acc;
#pragma unroll
    for (int j = 0; j < 8; ++j) acc[j] = 0.0f;
#pragma unroll
    for (int ks = 0; ks < 16; ++ks) {
      int kk = ks * 4 + khalf;
      v2f a;  a.x  = plane[(yt * 16 + ln15) * 68 + kk];
              a.y  = plane[(yt * 16 + ln15) * 68 + kk + 1];
      v2f bb; bb.x = dftz[kk * 16 + ln15];
              bb.y = dftz[(kk + 1) * 16 + ln15];
      acc = wmma_f32_k4(a, bb, acc);
    }
#pragma unroll
    for (int j = 0; j < 8; ++j) {
      int row = yt * 16 + j + ((lane >= 16) ? 8 : 0);
      zf[row * 17 + ln15] = acc[j];
    }
  }
  __syncthreads();

  // phase 2: forward Y-DFT, keep kh in {0..7, 56..63}; one output per thread
  {
    int kh_i = t >> 4, comp = t & 15, kz = comp >> 1;
    float accv = 0.0f;
    for (int y = 0; y < 64; ++y) {
      float a = zf[y * 17 + 2 * kz], b = zf[y * 17 + 2 * kz + 1];
      float cv = twy[(kh_i * 64 + y) * 2], sv = twy[(kh_i * 64 + y) * 2 + 1];
      accv += (comp & 1) ? (b * cv - a * sv) : (a * cv + b * sv);  // e^{-i th}
    }
    F12[(size_t)blk * 256 + t] = accv;
  }
}

// ---------------------------------------------------------------------------
// K2: forward X-DFT. F12[b][c][x64][256] -> F3[b][c][kd16][256]. Data stays
// L2-resident (16MB); re-read from global instead of staging 64KB in LDS.
// ---------------------------------------------------------------------------
__global__ __launch_bounds__(256) void k2_fwd_x(const float* __restrict__ F12,
                                                float* __restrict__ F3) {
  __shared__ float twx[16 * 64 * 2];
  const int t = threadIdx.x, blk = blockIdx.x;  // b*32+c
#pragma unroll
  for (int i = 0; i < 4; ++i) {
    int e = t * 4 + i;  // kd_i*64 + xx
    int kd_i = e >> 6, xx = e & 63;
    int kd = (kd_i < 8) ? kd_i : kd_i + 48;
    float th = (2.0f * PI_F / 64.0f) * (float)(kd * xx);
    float sv, cv; __sincosf(th, &sv, &cv);
    twx[e * 2] = cv; twx[e * 2 + 1] = sv;
  }
  __syncthreads();
  const int col = t & 127, half = t >> 7;
  const float* in = F12 + (size_t)blk * 16384;
  float* out = F3 + (size_t)blk * 4096;
  for (int kd_i = half * 8; kd_i < half * 8 + 8; ++kd_i) {
    float re = 0.0f, im = 0.0f;
    for (int xx = 0; xx < 64; ++xx) {
      float a = in[xx * 256 + 2 * col], b = in[xx * 256 + 2 * col + 1];
      float cv = twx[(kd_i * 64 + xx) * 2], sv = twx[(kd_i * 64 + xx) * 2 + 1];
      re += a * cv + b * sv;
      im += b * cv - a * sv;
    }
    out[kd_i * 256 + 2 * col] = re;
    out[kd_i * 256 + 2 * col + 1] = im;
  }
}

// ---------------------------------------------------------------------------
// K3: complex channel mix per mode. One block per (b, kd_i, kh_i); thread
// covers (o, kz). Weight tensor selected by spectral corner.
// ---------------------------------------------------------------------------
__global__ __launch_bounds__(256) void k3_chanmix(
    const float* __restrict__ F3, const float* __restrict__ w1,
    const float* __restrict__ w2, const float* __restrict__ w3,
    const float* __restrict__ w4, float* __restrict__ G) {
  __shared__ float xs[512];  // [i32][2kz+p]
  const int t = threadIdx.x;
  const int blk = blockIdx.x;  // b*256 + kd_i*16 + kh_i
  const int b = blk >> 8, kd_i = (blk >> 4) & 15, kh_i = blk & 15;
  for (int e = t; e < 512; e += 256) {
    int i = e >> 4, comp = e & 15;
    xs[e] = F3[((size_t)(b * 32 + i) * 16 + kd_i) * 256 + kh_i * 16 + comp];
  }
  __syncthreads();
  const float* w = (kd_i < 8) ? ((kh_i < 8) ? w1 : w3)
                              : ((kh_i < 8) ? w2 : w4);
  const int md = kd_i & 7, mh = kh_i & 7;
  const int o = t >> 3, kz = t & 7;
  float re = 0.0f, im = 0.0f;
  for (int i = 0; i < 32; ++i) {
    float a = xs[i * 16 + 2 * kz], bb = xs[i * 16 + 2 * kz + 1];
    size_t widx = ((size_t)(((i * 32 + o) * 8 + md) * 8 + mh) * 8 + kz) * 2;
    float wr = w[widx], wi = w[widx + 1];
    re += a * wr - bb * wi;   // real = ar*br - ai*bi
    im += bb * wr + a * wi;   // imag = ai*br + ar*bi
  }
  size_t oidx = ((size_t)(b * 32 + o) * 16 + kd_i) * 256 + kh_i * 16 + 2 * kz;
  G[oidx] = re;
  G[oidx + 1] = im;
}

// ---------------------------------------------------------------------------
// K4: inverse X (16 modes -> 64 points). G[b][o][kd16][256] -> G1[b][o][x64][256].
// ---------------------------------------------------------------------------
__global__ __launch_bounds__(256) void k4_inv_x(const float* __restrict__ G,
                                                float* __restrict__ G1) {
  __shared__ float twxi[16 * 64 * 2];
  const int t = threadIdx.x, blk = blockIdx.x;  // b*32+o
#pragma unroll
  for (int i = 0; i < 4; ++i) {
    int e = t * 4 + i;
    int kd_i = e >> 6, xx = e & 63;
    int kd = (kd_i < 8) ? kd_i : kd_i + 48;
    float th = (2.0f * PI_F / 64.0f) * (float)(kd * xx);
    float sv, cv; __sincosf(th, &sv, &cv);
    twxi[e * 2] = cv; twxi[e * 2 + 1] = sv;
  }
  __syncthreads();
  const int col = t & 127, half = t >> 7;
  const float* in = G + (size_t)blk * 4096;
  float* out = G1 + (size_t)blk * 16384;
  for (int xx = half * 32; xx < half * 32 + 32; ++xx) {
    float re = 0.0f, im = 0.0f;
    for (int kd_i = 0; kd_i < 16; ++kd_i) {
      float a = in[kd_i * 256 + 2 * col], b = in[kd_i * 256 + 2 * col + 1];
      float cv = twxi[(kd_i * 64 + xx) * 2], sv = twxi[(kd_i * 64 + xx) * 2 + 1];
      re += a * cv - b * sv;  // e^{+i th}
      im += a * sv + b * cv;
    }
    out[xx * 256 + 2 * col] = re;
    out[xx * 256 + 2 * col + 1] = im;
  }
}

// ---------------------------------------------------------------------------
// K5: fused inverse Y (VALU) + inverse real Z (WMMA) for one (b,o,x) plane.
// Inverse ortho 1/512 and Hermitian doubling folded into the 16x64 Z matrix:
//   x(z) = sum_kz s_kz * (a*cos - b*sin) / 512,  s_0=1, s_{1..7}=2.
// ---------------------------------------------------------------------------
__global__ __launch_bounds__(256) void k5_inv_yz(const float* __restrict__ G1,
                                                 float* __restrict__ out) {
  __shared__ __align__(16) float g1s[256];  // [kh16][2kz+p]
  __shared__ float yz[64 * 17];             // [y][2kz+p], padded
  __shared__ float idftz[16 * 64];          // [2kz+p][z]
  __shared__ float twyi[16 * 64 * 2];

  const int t = threadIdx.x;
  const int blk = blockIdx.x;  // (b*32+o)*64 + x

#if HAVE_ASYNC_LDS
  if (t < 64) {  // 256 floats = 64 x 16B async chunks
    v4i* g = (v4i*)(void*)const_cast<float*>(G1 + (size_t)blk * 256);
    __builtin_amdgcn_global_load_async_to_lds_b128(
        g + t, (v4i*)(void*)&g1s[t * 4], 0, 0);
  }
  __builtin_amdgcn_s_wait_asynccnt(0);
#else
  g1s[t] = G1[(size_t)blk * 256 + t];
#endif
#pragma unroll
  for (int i = 0; i < 4; ++i) {
    int e = t * 4 + i;  // comp*64 + z
    int comp = e >> 6, z = e & 63, kz = comp >> 1;
    float th = (2.0f * PI_F / 64.0f) * (float)(kz * z);
    float sv, cv; __sincosf(th, &sv, &cv);
    float s_k = (kz == 0) ? 1.0f : 2.0f;
    idftz[e] = ((comp & 1) ? -sv : cv) * s_k * (1.0f / 512.0f);
  }
#pragma unroll
  for (int i = 0; i < 4; ++i) {
    int e = t * 4 + i;  // kh_i*64 + y
    int kh_i = e >> 6, y = e & 63;
    int kh = (kh_i < 8) ? kh_i : kh_i + 48;
    float th = (2.0f * PI_F / 64.0f) * (float)(kh * y);
    float sv, cv; __sincosf(th, &sv, &cv);
    twyi[e * 2] = cv; twyi[e * 2 + 1] = sv;
  }
  __syncthreads();

  // inverse Y: yz[y][comp] = sum_kh g1s[kh][.] * e^{+i th}
#pragma unroll
  for (int i = 0; i < 4; ++i) {
    int e = t * 4 + i;  // y*16 + comp
    int y = e >> 4, comp = e & 15, kz = comp >> 1;
    float acc = 0.0f;
    for (int kh_i = 0; kh_i < 16; ++kh_i) {
      float a = g1s[kh_i * 16 + 2 * kz], b = g1s[kh_i * 16 + 2 * kz + 1];
      float cv = twyi[(kh_i * 64 + y) * 2], sv = twyi[(kh_i * 64 + y) * 2 + 1];
      acc += (comp & 1) ? (a * sv + b * cv) : (a * cv - b * sv);
    }
    yz[y * 17 + comp] = acc;
  }
  __syncthreads();

  // inverse Z via WMMA: out[64y][64z] = yz[64y][16] * idftz[16][64]
  const int wave = t >> 5, lane = t & 31;
  const int ln15 = lane & 15;
  const int khalf = (lane >> 4) * 2;
  float* obase = out + (size_t)blk * 4096;
#pragma unroll
  for (int rep = 0; rep < 2; ++rep) {
    int tile = wave + rep * 8;  // 16 tiles = 4 y-tiles x 4 z-tiles
    int yt = tile >> 2, zt = tile & 3;
    v8f acc;
#pragma unroll
    for (int j = 0; j < 8; ++j) acc[j] = 0.0f;
#pragma unroll
    for (int ks = 0; ks < 4; ++ks) {
      int kk = ks * 4 + khalf;
      v2f a;  a.x  = yz[(yt * 16 + ln15) * 17 + kk];
              a.y  = yz[(yt * 16 + ln15) * 17 + kk + 1];
      v2f bb; bb.x = idftz[kk * 64 + zt * 16 + ln15];
              bb.y = idftz[(kk + 1) * 64 + zt * 16 + ln15];
      acc = wmma_f32_k4(a, bb, acc);
    }
#pragma unroll
    for (int j = 0; j < 8; ++j) {
      int row = yt * 16 + j + ((lane >= 16) ? 8 : 0);
      obase[row * 64 + zt * 16 + ln15] = acc[j];
    }
  }
}

// ---------------------------------------------------------------------------
extern "C" void kernel_launch(void* const* d_in, const int* in_sizes, int n_in,
                              void* d_out, int out_size, void* d_ws,
                              size_t ws_size, hipStream_t stream) {
  const float* x  = (const float*)d_in[0];
  const float* w1 = (const float*)d_in[1];
  const float* w2 = (const float*)d_in[2];
  const float* w3 = (const float*)d_in[3];
  const float* w4 = (const float*)d_in[4];
  float* out = (float*)d_out;
  float* ws  = (float*)d_ws;

  float* F12 = ws;                 // [8][32][64][16][8][2] = 4,194,304 f
  float* F3  = F12 + 4194304;      // [8][32][16][16][8][2] = 1,048,576 f
  float* G   = F3 + 1048576;       // [8][32][16][16][8][2] = 1,048,576 f
  float* G1  = G + 1048576;        // [8][32][64][16][8][2] = 4,194,304 f
                                   // total 40 MiB of workspace

  k1_fwd_zy<<<8 * 32 * 64, 256, 0, stream>>>(x, F12);
  k2_fwd_x<<<8 * 32, 256, 0, stream>>>(F12, F3);
  k3_chanmix<<<8 * 16 * 16, 256, 0, stream>>>(F3, w1, w2, w3, w4, G);
  k4_inv_x<<<8 * 32, 256, 0, stream>>>(G, G1);
  k5_inv_yz<<<8 * 32 * 64, 256, 0, stream>>>(G1, out);
}